// SelectiveCentroidLayer_48610439856175
// MI455X (gfx1250) — compile-verified
//
#include <hip/hip_runtime.h>
#include <hip/hip_bf16.h>

// Problem constants (match reference)
#define B_ROWS 8192
#define C_CLS  1000
#define D_DIM  512
#define T_TOT  2000          // C*N
#define T_PAD  2048          // padded column count (16 steps of 128)
#define OC     1001          // output columns: 1000 class scores + 1 rej dist
#define LDSTR  516           // padded LDS row stride in floats (516 % 64 == 4 -> conflict-free)
#define INF_BITS 0x7F800000u

typedef float v2f __attribute__((ext_vector_type(2)));
typedef float v8f __attribute__((ext_vector_type(8)));

static __device__ __forceinline__ unsigned long long shfl_xor_u64(unsigned long long v, int m) {
    unsigned lo = (unsigned)(v & 0xffffffffull);
    unsigned hi = (unsigned)(v >> 32);
    lo = __shfl_xor(lo, m, 32);
    hi = __shfl_xor(hi, m, 32);
    return ((unsigned long long)hi << 32) | lo;
}

// ---------------------------------------------------------------------------
// K_init: zero the max-norm2 accumulator (ws is poisoned, not re-zeroed)
// ---------------------------------------------------------------------------
__global__ void k_init(unsigned* __restrict__ maxn2) {
    if (threadIdx.x == 0) *maxn2 = 0u;
}

// ---------------------------------------------------------------------------
// K_rownorm: norm2[t] for centroids (+ atomicMax of max norm2), r2[t] for rej.
// grid = 2*T blocks of 256 threads, one row each.
// ---------------------------------------------------------------------------
__global__ __launch_bounds__(256) void k_rownorm(const float* __restrict__ cent,
                                                 const float* __restrict__ rejc,
                                                 float* __restrict__ norm2,
                                                 float* __restrict__ r2,
                                                 unsigned* __restrict__ maxn2) {
    int row = blockIdx.x;
    const bool isRej = row >= T_TOT;
    if (isRej) row -= T_TOT;
    const float* src = (isRej ? rejc : cent) + (size_t)row * D_DIM;

    const int tid = threadIdx.x;
    float p = 0.f;
    for (int k = tid; k < D_DIM; k += 256) { float v = src[k]; p += v * v; }
#pragma unroll
    for (int m = 16; m >= 1; m >>= 1) p += __shfl_xor(p, m, 32);

    __shared__ float red[8];
    const int lane = tid & 31, wv = tid >> 5;
    if (lane == 0) red[wv] = p;
    __syncthreads();
    if (tid == 0) {
        float s = 0.f;
#pragma unroll
        for (int i = 0; i < 8; ++i) s += red[i];
        if (isRej) r2[row] = s;
        else {
            norm2[row] = s;
            atomicMax(maxn2, __float_as_uint(s));  // s >= 0 -> bit-monotone
        }
    }
}

// ---------------------------------------------------------------------------
// K_dist: fused  dist = sqrt(relu(x2 + c2 - 2*inv*(x@cent_raw.T)))
//   + per-class pair-min -> y, + per-row packed argmin -> minpack.
// Block: 256 thr (8 waves), 32 rows; wave covers 16 cols/step, 2 accumulators.
// ---------------------------------------------------------------------------
__global__ __launch_bounds__(256) void k_dist(const float* __restrict__ x,
                                              const float* __restrict__ cent,
                                              const float* __restrict__ norm2,
                                              const unsigned* __restrict__ maxn2,
                                              float* __restrict__ out,
                                              unsigned long long* __restrict__ minpack) {
    __shared__ float lds_x[32 * LDSTR];
    __shared__ float sh_x2[32];
    __shared__ unsigned long long sh_mp[32];

    const int tid  = threadIdx.x;
    const int lane = tid & 31;
    const int wv   = tid >> 5;
    const int half = lane >> 4;      // K-half for WMMA A/B operands
    const int lm   = lane & 15;      // A-row / B-col within tile
    const int koff = half * 2;
    const int rowbase = blockIdx.x * 32;

    // stage 32x512 x-tile in LDS (float4)
    for (int i = tid; i < 32 * (D_DIM / 4); i += 256) {
        const int m  = i / (D_DIM / 4);
        const int k4 = (i % (D_DIM / 4)) * 4;
        const float4 v = *(const float4*)(x + (size_t)(rowbase + m) * D_DIM + k4);
        *(float4*)(&lds_x[m * LDSTR + k4]) = v;
    }
    if (tid < 32) sh_mp[tid] = ~0ull;
    __syncthreads();
    if (tid < 32) {
        float s = 0.f;
        for (int k = 0; k < D_DIM; ++k) { float v = lds_x[tid * LDSTR + k]; s += v * v; }
        sh_x2[tid] = s;
    }
    __syncthreads();

    const float cn   = sqrtf(__uint_as_float(*maxn2)) + 1e-16f;
    const float inv  = 1.f / cn;
    const float inv2 = inv * inv;

    for (int tbase = 0; tbase < T_PAD; tbase += 128) {
        const int  tcol   = tbase + (wv << 4) + lm;
        const int  tcl    = tcol < T_TOT ? tcol : (T_TOT - 1);
        const bool tvalid = tcol < T_TOT;
        const float* brow = cent + (size_t)tcl * D_DIM;

        v8f accA = {0.f, 0.f, 0.f, 0.f, 0.f, 0.f, 0.f, 0.f};
        v8f accB = {0.f, 0.f, 0.f, 0.f, 0.f, 0.f, 0.f, 0.f};
#pragma unroll 4
        for (int k = 0; k < D_DIM; k += 4) {
            v2f a0 = *(const v2f*)(&lds_x[lm * LDSTR + k + koff]);
            v2f a1 = *(const v2f*)(&lds_x[(16 + lm) * LDSTR + k + koff]);
            v2f b  = *(const v2f*)(brow + k + koff);
            accA = __builtin_amdgcn_wmma_f32_16x16x4_f32(false, a0, false, b,
                                                         (short)0, accA, false, false);
            accB = __builtin_amdgcn_wmma_f32_16x16x4_f32(false, a1, false, b,
                                                         (short)0, accB, false, false);
        }

        const float c2 = norm2[tcl] * inv2;
#pragma unroll
        for (int vr = 0; vr < 8; ++vr) {
#pragma unroll
            for (int g = 0; g < 2; ++g) {
                const int   m = g * 16 + vr + 8 * half;   // row within 32-row tile
                const float P = (g == 0) ? accA[vr] : accB[vr];
                const float d2   = sh_x2[m] + c2 - 2.f * inv * P;
                const float dist = tvalid ? sqrtf(fmaxf(d2, 0.f)) : __uint_as_float(INF_BITS);

                // class pair min: lanes (2j, 2j+1) hold (t, t+1) of one class
                const float other = __shfl_xor(dist, 1, 32);
                if (((lane & 1) == 0) && tvalid) {
                    out[(size_t)(rowbase + m) * OC + (tcol >> 1)] = -fminf(dist, other);
                }

                // global argmin over t (lexicographic (dist, idx) min == first argmin)
                unsigned long long pk =
                    ((unsigned long long)__float_as_uint(dist) << 32) | (unsigned)tcl;
#pragma unroll
                for (int msk = 1; msk <= 8; msk <<= 1) {
                    const unsigned long long o = shfl_xor_u64(pk, msk);
                    if (o < pk) pk = o;
                }
                if (lm == 0) atomicMin(&sh_mp[m], pk);
            }
        }
    }
    __syncthreads();
    if (tid < 32) minpack[rowbase + tid] = sh_mp[tid];
}

// ---------------------------------------------------------------------------
// K_rej: gather closest = inv*cent[idx]; XR = x@r.T, CR = closest@r.T (WMMA);
//   rej[b] = min_t |cx - XR - 0.5*(cl2 - r2)| / (sqrt(relu(cl2 + r2 - 2CR)) + 1e-9)
// Block: 256 thr (8 waves), 16 rows.
// ---------------------------------------------------------------------------
__global__ __launch_bounds__(256) void k_rej(const float* __restrict__ x,
                                             const float* __restrict__ cent,
                                             const float* __restrict__ rejc,
                                             const float* __restrict__ norm2,
                                             const float* __restrict__ r2,
                                             const unsigned* __restrict__ maxn2,
                                             const unsigned long long* __restrict__ minpack,
                                             float* __restrict__ out) {
    __shared__ float lds_x[16 * LDSTR];
    __shared__ float lds_c[16 * LDSTR];
    __shared__ float sh_cx[16];
    __shared__ float sh_cl2[16];
    __shared__ unsigned sh_rej[16];
    __shared__ unsigned sh_idx[16];

    const int tid  = threadIdx.x;
    const int lane = tid & 31;
    const int wv   = tid >> 5;
    const int half = lane >> 4;
    const int lm   = lane & 15;
    const int koff = half * 2;
    const int rowbase = blockIdx.x * 16;

    const float cn   = sqrtf(__uint_as_float(*maxn2)) + 1e-16f;
    const float inv  = 1.f / cn;
    const float inv2 = inv * inv;

    if (tid < 16) {
        const unsigned long long pk = minpack[rowbase + tid];
        sh_idx[tid] = (unsigned)(pk & 0xffffffffull);
        sh_rej[tid] = INF_BITS;
    }
    __syncthreads();

    for (int i = tid; i < 16 * (D_DIM / 4); i += 256) {
        const int m  = i / (D_DIM / 4);
        const int k4 = (i % (D_DIM / 4)) * 4;
        const float4 vx = *(const float4*)(x + (size_t)(rowbase + m) * D_DIM + k4);
        float4 vc = *(const float4*)(cent + (size_t)sh_idx[m] * D_DIM + k4);
        vc.x *= inv; vc.y *= inv; vc.z *= inv; vc.w *= inv;
        *(float4*)(&lds_x[m * LDSTR + k4]) = vx;
        *(float4*)(&lds_c[m * LDSTR + k4]) = vc;
    }
    __syncthreads();
    if (tid < 16) {
        float s = 0.f;
        for (int k = 0; k < D_DIM; ++k) s += lds_c[tid * LDSTR + k] * lds_x[tid * LDSTR + k];
        sh_cx[tid]  = s;
        sh_cl2[tid] = norm2[sh_idx[tid]] * inv2;
    }
    __syncthreads();

    for (int tbase = 0; tbase < T_PAD; tbase += 128) {
        const int  tcol   = tbase + (wv << 4) + lm;
        const int  tcl    = tcol < T_TOT ? tcol : (T_TOT - 1);
        const bool tvalid = tcol < T_TOT;
        const float* brow = rejc + (size_t)tcl * D_DIM;

        v8f accX = {0.f, 0.f, 0.f, 0.f, 0.f, 0.f, 0.f, 0.f};   // x @ r.T
        v8f accC = {0.f, 0.f, 0.f, 0.f, 0.f, 0.f, 0.f, 0.f};   // closest @ r.T
#pragma unroll 4
        for (int k = 0; k < D_DIM; k += 4) {
            v2f ax = *(const v2f*)(&lds_x[lm * LDSTR + k + koff]);
            v2f ac = *(const v2f*)(&lds_c[lm * LDSTR + k + koff]);
            v2f b  = *(const v2f*)(brow + k + koff);
            accX = __builtin_amdgcn_wmma_f32_16x16x4_f32(false, ax, false, b,
                                                         (short)0, accX, false, false);
            accC = __builtin_amdgcn_wmma_f32_16x16x4_f32(false, ac, false, b,
                                                         (short)0, accC, false, false);
        }

        const float r2t = r2[tcl];
#pragma unroll
        for (int vr = 0; vr < 8; ++vr) {
            const int   m   = vr + 8 * half;
            const float xr  = accX[vr];
            const float cr  = accC[vr];
            const float num = fabsf(sh_cx[m] - xr - 0.5f * (sh_cl2[m] - r2t));
            const float den = sqrtf(fmaxf(sh_cl2[m] + r2t - 2.f * cr, 0.f)) + 1e-9f;
            float ratio = tvalid ? (num / den) : __uint_as_float(INF_BITS);
#pragma unroll
            for (int msk = 1; msk <= 8; msk <<= 1)
                ratio = fminf(ratio, __shfl_xor(ratio, msk, 32));
            if (lm == 0) atomicMin(&sh_rej[m], __float_as_uint(ratio));
        }
    }
    __syncthreads();
    if (tid < 16)
        out[(size_t)(rowbase + tid) * OC + C_CLS] = __uint_as_float(sh_rej[tid]);
}

// ---------------------------------------------------------------------------
extern "C" void kernel_launch(void* const* d_in, const int* in_sizes, int n_in,
                              void* d_out, int out_size, void* d_ws, size_t ws_size,
                              hipStream_t stream) {
    const float* x    = (const float*)d_in[0];   // [B, D]
    const float* cent = (const float*)d_in[1];   // [C, N, D] -> [T, D]
    const float* rejc = (const float*)d_in[2];   // [T, D]
    // d_in[3] = return_dists (always 1 in this setup)
    float* out = (float*)d_out;                  // [B, 1001]

    // workspace layout (floats): norm2[T] | r2[T] | maxn2 (u32) | pad | minpack[B] (u64)
    float* ws = (float*)d_ws;
    float*    ws_norm2 = ws;
    float*    ws_r2    = ws + T_TOT;
    unsigned* ws_maxn2 = (unsigned*)(ws + 2 * T_TOT);
    unsigned long long* ws_minpack = (unsigned long long*)(ws + 2 * T_TOT + 4); // 8B aligned

    hipLaunchKernelGGL(k_init, dim3(1), dim3(64), 0, stream, ws_maxn2);
    hipLaunchKernelGGL(k_rownorm, dim3(2 * T_TOT), dim3(256), 0, stream,
                       cent, rejc, ws_norm2, ws_r2, ws_maxn2);
    hipLaunchKernelGGL(k_dist, dim3(B_ROWS / 32), dim3(256), 0, stream,
                       x, cent, ws_norm2, ws_maxn2, out, ws_minpack);
    hipLaunchKernelGGL(k_rej, dim3(B_ROWS / 16), dim3(256), 0, stream,
                       x, cent, rejc, ws_norm2, ws_r2, ws_maxn2, ws_minpack, out);
}